// LogSig_var_1614907703723
// MI455X (gfx1250) — compile-verified
//
#include <hip/hip_runtime.h>

// Depth-2 log-signature (N_SEGMENTS=64) of ragged paths, B=256, T=2048, d=12.
// Memory-bound (~30MB traffic -> ~1.3us floor at 23.3TB/s). Per-segment outer-
// product accumulation Sum_p Xe[p] (x) dX[p] is a 12x12 GEMM over K=len, mapped
// to V_WMMA_F32_16X16X4_F32. A second WMMA with swapped operands accumulates
// the transpose so the Levy antisymmetrization is purely in-lane.
//
// Loop bounds are forced into SGPRs (readfirstlane) so the chunk loop is a
// scalar-controlled loop: EXEC is guaranteed all-ones at every WMMA and loop
// overhead is SALU-only. The common k==1 (non-expand) path has no integer
// division; the rare expand path (L<65, Leff<=128) keeps it.

typedef __attribute__((ext_vector_type(2))) float v2f;
typedef __attribute__((ext_vector_type(8))) float v8f;

#define DCH 12
#define NSEG 64

__global__ __launch_bounds__(256) void logsig_var_kernel(
    const float* __restrict__ X,      // (B, T, 12)
    const int*   __restrict__ length, // (B,)
    float*       __restrict__ out,    // (B, 64, 78)
    int T)
{
  const int b    = blockIdx.x;
  const int lane = threadIdx.x & 31;
  const int wave = threadIdx.x >> 5;
  const int g    = lane >> 4;   // lane half: selects K pair within chunk
  const int f    = lane & 15;   // feature index (A row / B,C,D column)

  const float* __restrict__ Xb = X + (size_t)b * T * DCH;
  // Pad lanes (f>=12) alias feature 0: their garbage only reaches D rows/cols
  // >=12, which are never read out.
  const float* __restrict__ XbF = Xb + (f < DCH ? f : 0);

  // Wave-uniform scalars, forced into SGPRs.
  const int  L      = __builtin_amdgcn_readfirstlane(length[b]);
  const bool expand = (L < NSEG + 1);
  const int  k      = expand ? (NSEG / L + 1) : 1;
  const int  Leff   = expand ? (k * L) : L;
  const float leff1 = (float)(Leff - 1);
  const unsigned uk = (unsigned)k;
  const int  Lm1    = L - 1;

  float* __restrict__ outb = out + (size_t)b * NSEG * 78;

  for (int s = 0; s < 8; ++s) {
    const int seg = wave * 8 + s;
    // tv = round(1 + j*(Leff-1)/64); rintf = round-half-to-even == jnp.round
    const int a  = __builtin_amdgcn_readfirstlane(
        (int)rintf(1.0f + (float)seg       * leff1 * 0.015625f) - 1);
    const int bb = __builtin_amdgcn_readfirstlane(
        (int)rintf(1.0f + (float)(seg + 1) * leff1 * 0.015625f) - 1);

    int ia, ib;
    if (k == 1) {
      ia = (a  < Lm1) ? a  : Lm1;
      ib = (bb < Lm1) ? bb : Lm1;
    } else {
      ia = (int)((unsigned)a  / uk); ia = (ia < Lm1) ? ia : Lm1;
      ib = (int)((unsigned)bb / uk); ib = (ib < Lm1) ? ib : Lm1;
    }
    __builtin_prefetch(Xb + (size_t)ia * DCH, 0, 3);

    v8f C1 = {}; // Sum_p Xe[p] (x) dX[p]   (rows=i, cols=j)
    v8f C2 = {}; // its transpose, via swapped operands

    if (k == 1) {
      // Fast path: idx(p) = min(p, L-1), no division.
      for (int p0 = a; p0 < bb; p0 += 4) {
        const int pa0 = p0 + 2 * g;
        const int i0 = (pa0     < Lm1) ? pa0     : Lm1;
        const int i1 = (pa0 + 1 < Lm1) ? pa0 + 1 : Lm1;
        const int i2 = (pa0 + 2 < Lm1) ? pa0 + 2 : Lm1;
        const float x0 = XbF[(size_t)i0 * DCH];
        const float x1 = XbF[(size_t)i1 * DCH];
        const float x2 = XbF[(size_t)i2 * DCH];
        const float d0 = (pa0     < bb) ? (x1 - x0) : 0.0f;  // K-tail mask
        const float d1 = (pa0 + 1 < bb) ? (x2 - x1) : 0.0f;

        v2f Aop = {x0, x1};   // A 16x4: (M=f, K=p0+2g / p0+2g+1)
        v2f Bop = {d0, d1};   // B 4x16: (N=f, same K mapping)
        C1 = __builtin_amdgcn_wmma_f32_16x16x4_f32(
                false, Aop, false, Bop, (short)0, C1, false, false);
        C2 = __builtin_amdgcn_wmma_f32_16x16x4_f32(
                false, Bop, false, Aop, (short)0, C2, false, false);
      }
    } else {
      // Expand path (rare, L<65, Leff<=128): idx(p) = min(p/k, L-1).
      for (int p0 = a; p0 < bb; p0 += 4) {
        const int pa0 = p0 + 2 * g;
        int i0 = (int)((unsigned)(pa0    ) / uk); i0 = (i0 < Lm1) ? i0 : Lm1;
        int i1 = (int)((unsigned)(pa0 + 1) / uk); i1 = (i1 < Lm1) ? i1 : Lm1;
        int i2 = (int)((unsigned)(pa0 + 2) / uk); i2 = (i2 < Lm1) ? i2 : Lm1;
        const float x0 = XbF[(size_t)i0 * DCH];
        const float x1 = XbF[(size_t)i1 * DCH];
        const float x2 = XbF[(size_t)i2 * DCH];
        const float d0 = (pa0     < bb) ? (x1 - x0) : 0.0f;
        const float d1 = (pa0 + 1 < bb) ? (x2 - x1) : 0.0f;

        v2f Aop = {x0, x1};
        v2f Bop = {d0, d1};
        C1 = __builtin_amdgcn_wmma_f32_16x16x4_f32(
                false, Aop, false, Bop, (short)0, C1, false, false);
        C2 = __builtin_amdgcn_wmma_f32_16x16x4_f32(
                false, Bop, false, Aop, (short)0, C2, false, false);
      }
    }

    // Segment endpoints (per-lane column values)
    const float xa_c  = (f < DCH) ? Xb[(size_t)ia * DCH + f] : 0.0f;
    const float xb_c  = (f < DCH) ? Xb[(size_t)ib * DCH + f] : 0.0f;
    const float inc_c = xb_c - xa_c;

    float* __restrict__ os = outb + (size_t)seg * 78;

    // increments: channels 0..11
    if (g == 0 && f < DCH) os[f] = inc_c;

    // Levy areas: channels 12..77; element (row,col) lives in lane(col,g), VGPR r
#pragma unroll
    for (int r = 0; r < 8; ++r) {
      const int row = r + 8 * g;
      if (row < f && f < DCH) {            // upper triangle, row < col < 12
        const float xar  = Xb[(size_t)ia * DCH + row];
        const float xbr  = Xb[(size_t)ib * DCH + row];
        const float incr = xbr - xar;
        const float gs   = C1[r] - xar  * inc_c;  // Gs[row][col]
        const float gst  = C2[r] - xa_c * incr;   // Gs[col][row]
        const float aval = 0.5f * (gs - gst);
        const int ch = 12 + row * 11 - (row * (row - 1)) / 2 + (f - row - 1);
        os[ch] = aval;
      }
    }
  }
}

extern "C" void kernel_launch(void* const* d_in, const int* in_sizes, int n_in,
                              void* d_out, int out_size, void* d_ws, size_t ws_size,
                              hipStream_t stream) {
  const float* X   = (const float*)d_in[0];
  const int*   len = (const int*)d_in[1];
  float*       out = (float*)d_out;

  const int B = in_sizes[1];                 // 256
  const int T = in_sizes[0] / (B * DCH);     // 2048

  logsig_var_kernel<<<B, 256, 0, stream>>>(X, len, out, T);
}